// CorrelationHead_45208825757816
// MI455X (gfx1250) — compile-verified
//
#include <hip/hip_runtime.h>

// MI455X / gfx1250, wave32. Fully fp32 pipeline using V_WMMA_F32_16X16X4_F32.
//
// Key algorithmic optimization: the spatial-correlation output (patch 16, dil 2,
// 7x7 map) is structurally sparse -- only 625 of 12544 flattened features can be
// nonzero (patch offsets pi,pj in [5,11] with clipped windows). We compact both
// the correlation output and the matching columns of W1, shrinking GEMM1's K
// from 12544 to 625 (padded 640). After that the whole problem is HBM-read
// bound (~158 MB ~ 6.8 us at 23.3 TB/s) and fp32 WMMA is more than enough.
//
// GEMM double-buffers its LDS tiles and uses GLOBAL_LOAD_ASYNC_TO_LDS_B128 +
// s_wait_asynccnt to overlap global-memory latency with WMMA compute.

typedef float v2f __attribute__((ext_vector_type(2)));
typedef float v8f __attribute__((ext_vector_type(8)));
typedef int   v4i __attribute__((vector_size(16)));   // matches builtin proto

#define NB    1024
#define NC    256
#define NHW   49
#define NFLAT 12544
#define NREP  1024
#define NQ    625
#define NQP   640   // NQ padded to a multiple of 16 (zero-filled)

#if defined(__has_builtin)
#if __has_builtin(__builtin_amdgcn_global_load_async_to_lds_b128)
#define HAVE_ASYNC_LDS 1
#endif
#endif
#ifndef HAVE_ASYNC_LDS
#define HAVE_ASYNC_LDS 0
#endif

#if HAVE_ASYNC_LDS
typedef __attribute__((address_space(1))) v4i as1_v4i;
typedef __attribute__((address_space(3))) v4i as3_v4i;

template <int N>
__device__ __forceinline__ void wait_asynccnt() {
#if __has_builtin(__builtin_amdgcn_s_wait_asynccnt)
  __builtin_amdgcn_s_wait_asynccnt(N);
#else
  asm volatile("s_wait_asynccnt %0" ::"i"(N) : "memory");
#endif
}
#endif

// ---------------------------------------------------------------------------
// Kernel 1: build the sparse-column tables (deterministic, rebuilt each call).
//   map[k_full]  -> q index in [0,625) or -1
//   qtab[q]      -> (ij = i*7+j,  uv = shifted index into p2 plane)
// ---------------------------------------------------------------------------
__global__ __launch_bounds__(256) void k_init_tables(int* __restrict__ map,
                                                     int2* __restrict__ qtab) {
  const int t = threadIdx.x;
  for (int k = t; k < NFLAT; k += 256) map[k] = -1;
  __syncthreads();
  if (t == 0) {
    int q = 0;
    for (int pi = 5; pi <= 11; ++pi) {
      int ilo = 16 - 2 * pi; if (ilo < 0) ilo = 0;
      int ihi = 22 - 2 * pi; if (ihi > 6) ihi = 6;
      for (int i = ilo; i <= ihi; ++i) {
        for (int pj = 5; pj <= 11; ++pj) {
          int jlo = 16 - 2 * pj; if (jlo < 0) jlo = 0;
          int jhi = 22 - 2 * pj; if (jhi > 6) jhi = 6;
          for (int j = jlo; j <= jhi; ++j) {
            int kfull = (pi * 16 + pj) * NHW + i * 7 + j;
            map[kfull] = q;
            qtab[q] = make_int2(i * 7 + j,
                                (i + 2 * pi - 16) * 7 + (j + 2 * pj - 16));
            ++q;
          }
        }
      }
    }
    // q == 625 here by construction (25 x 25 valid (pi,i)x(pj,j) pairs)
  }
}

// ---------------------------------------------------------------------------
// Kernel 2: sparse correlation -> Xc[b, q], q in [0,640) (tail zero-filled).
// One block per batch element; channels tiled through LDS in chunks of 64.
// Exact fp32; this stage is bound by the 103 MB read of p1/p2.
// ---------------------------------------------------------------------------
#define CCHUNK 64
__global__ __launch_bounds__(256) void k_corr(const float* __restrict__ p1,
                                              const float* __restrict__ p2,
                                              const int2* __restrict__ qtab,
                                              float* __restrict__ Xc) {
  __shared__ float s1[CCHUNK * NHW];
  __shared__ float s2[CCHUNK * NHW];
  const int b = blockIdx.x;
  const int t = threadIdx.x;

  int ij[3], uv[3];
  float acc[3];
#pragma unroll
  for (int r = 0; r < 3; ++r) {
    const int q = t + 256 * r;
    if (q < NQ) { int2 e = qtab[q]; ij[r] = e.x; uv[r] = e.y; }
    else        { ij[r] = 0;        uv[r] = 0; }
    acc[r] = 0.0f;
  }

  const size_t base = (size_t)b * NC * NHW;
  for (int c0 = 0; c0 < NC; c0 += CCHUNK) {
    for (int f = t; f < CCHUNK * NHW; f += 256) {
      s1[f] = p1[base + (size_t)c0 * NHW + f];
      s2[f] = p2[base + (size_t)c0 * NHW + f];
    }
    __syncthreads();
    for (int c = 0; c < CCHUNK; ++c) {
      const int o = c * NHW;
#pragma unroll
      for (int r = 0; r < 3; ++r)
        acc[r] = fmaf(s1[o + ij[r]], s2[o + uv[r]], acc[r]);
    }
    __syncthreads();
  }

#pragma unroll
  for (int r = 0; r < 3; ++r) {
    const int q = t + 256 * r;
    if (q < NQ)       Xc[(size_t)b * NQP + q] = acc[r];
    else if (q < NQP) Xc[(size_t)b * NQP + q] = 0.0f;
  }
}

// ---------------------------------------------------------------------------
// Kernel 3: compact W1 [1024 x 12544] -> W1c [1024 x 640] (coalesced stream).
// ---------------------------------------------------------------------------
__global__ __launch_bounds__(256) void k_compact_w1(const float* __restrict__ W1,
                                                    const int* __restrict__ map,
                                                    float* __restrict__ W1c) {
  const int n = blockIdx.x;
  const int t = threadIdx.x;
  if (t < NQP - NQ) W1c[(size_t)n * NQP + NQ + t] = 0.0f;
  for (int k = t; k < NFLAT; k += 256) {
    const float v = W1[(size_t)n * NFLAT + k];
    const int q = map[k];
    if (q >= 0) W1c[(size_t)n * NQP + q] = v;
  }
}

// ---------------------------------------------------------------------------
// Kernel 4/5: fp32 WMMA GEMM:  out[M x N] = act(A[M x K] * Bw[N x K]^T + bias)
// Block = 128 threads (4 waves), tile 64(M) x 64(N); wave -> 16 M-rows x 64 N.
// K stepped by 16 through double-buffered LDS (pitch 20 floats: conflict-free,
// 16B-aligned rows). Stage s+1 is loaded (async-to-LDS when available) while
// stage s is consumed by WMMA.
// WMMA f32 16x16x4 fragments:
//   A: lane L -> (m = L%16, k = v + 2*(L/16)),  v in {0,1}
//   B: lane L -> (k = v + 2*(L/16), n = L%16)
//   D: lane L, vgpr v -> (m = v + 8*(L/16), n = L%16)
// ---------------------------------------------------------------------------
#define TILE   64
#define PITCH  20
__global__ __launch_bounds__(128) void k_gemm_bias_act(
    const float* __restrict__ A,    // [M x K] row-major
    const float* __restrict__ Bw,   // [N x K] row-major (weights)
    const float* __restrict__ bias, // [N]
    float* __restrict__ out,        // [M x N] row-major
    int N, int K, int do_relu) {
  __shared__ float Alds[2][TILE * PITCH];
  __shared__ float Blds[2][TILE * PITCH];

  const int t    = threadIdx.x;
  const int wave = t >> 5;
  const int lane = t & 31;
  const int l16  = lane & 15;
  const int hi   = lane >> 4;   // 0 or 1
  const int m0   = blockIdx.y * TILE;
  const int n0   = blockIdx.x * TILE;

  // Branch-free cooperative tile load: thread t covers float4 slots t, t+128.
  const int row0 = t >> 2;            // rows 0..31 (and +32 for second slot)
  const int c4   = (t & 3) << 2;      // k-subcolumn 0/4/8/12

  v8f acc[4];
#pragma unroll
  for (int nt = 0; nt < 4; ++nt) {
    v8f z = {0.f, 0.f, 0.f, 0.f, 0.f, 0.f, 0.f, 0.f};
    acc[nt] = z;
  }

  auto issue_stage = [&](int k0, int buf) {
#pragma unroll
    for (int ff = 0; ff < 2; ++ff) {
      const int row = row0 + 32 * ff;
      const float* ga = A  + (size_t)(m0 + row) * K + k0 + c4;
      const float* gb = Bw + (size_t)(n0 + row) * K + k0 + c4;
      float* la = &Alds[buf][row * PITCH + c4];
      float* lb = &Blds[buf][row * PITCH + c4];
#if HAVE_ASYNC_LDS
      __builtin_amdgcn_global_load_async_to_lds_b128((as1_v4i*)ga, (as3_v4i*)la,
                                                     0, 0);
      __builtin_amdgcn_global_load_async_to_lds_b128((as1_v4i*)gb, (as3_v4i*)lb,
                                                     0, 0);
#else
      *(float4*)la = *(const float4*)ga;
      *(float4*)lb = *(const float4*)gb;
#endif
    }
  };

  int buf = 0;
  issue_stage(0, 0);
  for (int k0 = 0; k0 < K; k0 += 16) {
    const bool has_next = (k0 + 16) < K;
    if (has_next) issue_stage(k0 + 16, buf ^ 1);
#if HAVE_ASYNC_LDS
    // In-order async completion: <=4 outstanding means the 4 loads of the
    // current stage have landed in LDS (the newest 4 may still be in flight).
    if (has_next) wait_asynccnt<4>(); else wait_asynccnt<0>();
#endif
    __syncthreads();   // all threads' stage-k0 data visible

#pragma unroll
    for (int kk = 0; kk < 4; ++kk) {
      const int kb = (kk << 2) + (hi << 1);
      v2f a;
      a.x = Alds[buf][(wave * 16 + l16) * PITCH + kb];
      a.y = Alds[buf][(wave * 16 + l16) * PITCH + kb + 1];
#pragma unroll
      for (int nt = 0; nt < 4; ++nt) {
        v2f b;
        b.x = Blds[buf][(nt * 16 + l16) * PITCH + kb];
        b.y = Blds[buf][(nt * 16 + l16) * PITCH + kb + 1];
        acc[nt] = __builtin_amdgcn_wmma_f32_16x16x4_f32(
            false, a, false, b, (short)0, acc[nt], false, false);
      }
    }
    __syncthreads();   // done reading buf before it is refilled next stage
    buf ^= 1;
  }

  // Epilogue: bias + optional ReLU, per the 16x16 f32 C/D layout.
#pragma unroll
  for (int nt = 0; nt < 4; ++nt) {
    const int n = n0 + nt * 16 + l16;
    const float bv = bias[n];
#pragma unroll
    for (int v = 0; v < 8; ++v) {
      const int m = m0 + wave * 16 + v + 8 * hi;
      float val = acc[nt][v] + bv;
      if (do_relu) val = fmaxf(val, 0.0f);
      out[(size_t)m * N + n] = val;
    }
  }
}

// ---------------------------------------------------------------------------
// Kernel 6: head GEMM, out[b, 0..3] = H2[b,:] . W3[o,:] + b3[o]
// ---------------------------------------------------------------------------
__global__ __launch_bounds__(128) void k_head(const float* __restrict__ H2,
                                              const float* __restrict__ W3,
                                              const float* __restrict__ b3,
                                              float* __restrict__ out) {
  __shared__ float red[4 * 128];
  const int b = blockIdx.x;
  const int t = threadIdx.x;
  float s[4] = {0.f, 0.f, 0.f, 0.f};
  for (int k = t; k < NREP; k += 128) {
    const float h = H2[(size_t)b * NREP + k];
#pragma unroll
    for (int o = 0; o < 4; ++o) s[o] = fmaf(h, W3[o * NREP + k], s[o]);
  }
#pragma unroll
  for (int o = 0; o < 4; ++o) red[o * 128 + t] = s[o];
  __syncthreads();
  for (int st = 64; st > 0; st >>= 1) {
    if (t < st) {
#pragma unroll
      for (int o = 0; o < 4; ++o) red[o * 128 + t] += red[o * 128 + t + st];
    }
    __syncthreads();
  }
  if (t < 4) out[b * 4 + t] = red[t * 128] + b3[t];
}

// ---------------------------------------------------------------------------
// Launch. Inputs (setup_inputs order): patch1, patch2, W1, b1, W2, b2, W3, b3.
// Workspace layout (needs ~13.7 MB):
//   [0)        map   : 12544 int
//   [50432)    qtab  : 625 int2
//   [65536)    Xc    : 1024*640 f32
//   [+2621440) W1c   : 1024*640 f32
//   [+2621440) H1    : 1024*1024 f32
//   [+4194304) H2    : 1024*1024 f32
// ---------------------------------------------------------------------------
extern "C" void kernel_launch(void* const* d_in, const int* in_sizes, int n_in,
                              void* d_out, int out_size, void* d_ws, size_t ws_size,
                              hipStream_t stream) {
  const float* p1 = (const float*)d_in[0];
  const float* p2 = (const float*)d_in[1];
  const float* W1 = (const float*)d_in[2];
  const float* b1 = (const float*)d_in[3];
  const float* W2 = (const float*)d_in[4];
  const float* b2 = (const float*)d_in[5];
  const float* W3 = (const float*)d_in[6];
  const float* b3 = (const float*)d_in[7];
  float* out = (float*)d_out;

  char* ws = (char*)d_ws;
  int*  map  = (int*)(ws);
  int2* qtab = (int2*)(ws + 50432);
  float* Xc  = (float*)(ws + 65536);
  float* W1c = Xc  + (size_t)NB * NQP;
  float* H1  = W1c + (size_t)NREP * NQP;
  float* H2  = H1  + (size_t)NB * NREP;

  k_init_tables<<<1, 256, 0, stream>>>(map, qtab);
  k_corr<<<NB, 256, 0, stream>>>(p1, p2, qtab, Xc);
  k_compact_w1<<<NREP, 256, 0, stream>>>(W1, map, W1c);
  k_gemm_bias_act<<<dim3(NREP / TILE, NB / TILE), 128, 0, stream>>>(
      Xc, W1c, b1, H1, NREP, NQP, 1);
  k_gemm_bias_act<<<dim3(NREP / TILE, NB / TILE), 128, 0, stream>>>(
      H1, W2, b2, H2, NREP, NREP, 1);
  k_head<<<NB, 128, 0, stream>>>(H2, W3, b3, out);
}